// HashTableVoxelizedGaussianAdapterModule_86990267613197
// MI455X (gfx1250) — compile-verified
//
#include <hip/hip_runtime.h>
#include <math.h>
#include <stdint.h>

#ifndef TABLE_SIZE_C
#define TABLE_SIZE_C 4194304u           // 2^22
#endif
#define REDUCE_N   (3u * TABLE_SIZE_C)  // rows 0..2 of the hash table (12,582,912 floats)
#define RED_GRID   512u                 // reduce grid (blocks)
#define RED_BLOCK  256u                 // reduce block (threads)
#define RED_TILE   1024u                // floats per tile = 256 lanes * 4 floats (b128 each)
#define RED_PER_BLOCK (REDUCE_N / RED_GRID)        // 24576 floats
#define RED_NTILES    (RED_PER_BLOCK / RED_TILE)   // 24 tiles

// Native clang vector types.
typedef float v4f __attribute__((ext_vector_type(4)));
typedef int   v4i __attribute__((ext_vector_type(4)));
typedef __attribute__((address_space(3))) v4i lds_v4i;   // LDS-resident int4

// ---- gfx1250 async global->LDS path (probe via __has_builtin) -------------
#if defined(__has_builtin)
#if __has_builtin(__builtin_amdgcn_global_load_async_to_lds_b128) && \
    __has_builtin(__builtin_amdgcn_s_wait_asynccnt)
#define HTVG_ASYNC 1
#endif
#endif

#ifdef HTVG_ASYNC
// Builtin signature (from hipcc diagnostics):
//   (int4* global_src, __shared__ int4* lds_dst, imm offset, imm cpol)
// Cast through uintptr_t to drop const/qualifiers; for LDS the low 32 bits of
// the generic pointer are the LDS offset (flat LDS aperture mapping).
#define HTVG_TO_GBL(p) ((v4i*)(uintptr_t)(p))
#define HTVG_TO_LDS(p) ((lds_v4i*)(unsigned)(uintptr_t)(p))
#endif

// ---------------------------------------------------------------------------
// Kernel 0: zero the two f64 accumulators in workspace (graph-replay safe).
// ---------------------------------------------------------------------------
__global__ void htvg_zero_ws(double* __restrict__ ws) {
    if (blockIdx.x == 0 && threadIdx.x == 0) {
        ws[0] = 0.0;   // sum
        ws[1] = 0.0;   // sum of squares
    }
}

// ---------------------------------------------------------------------------
// Kernel 1: global mean / sum-of-squares over hash_table rows 0..2.
// Async double-buffered global->LDS streaming on gfx1250; each lane copies a
// 16B chunk and reads back only its own chunk, so s_wait_asynccnt (own-wave
// ordering) is the only synchronization needed inside the stream loop.
// ---------------------------------------------------------------------------
__global__ __launch_bounds__(RED_BLOCK) void htvg_reduce(const float* __restrict__ ht,
                                                         double* __restrict__ ws) {
    __shared__ float  tile[2][RED_TILE];     // 2 x 4KB double buffer
    __shared__ double s_sum[RED_BLOCK];
    __shared__ double s_sq[RED_BLOCK];

    const unsigned tid  = threadIdx.x;
    const unsigned base = blockIdx.x * RED_PER_BLOCK;   // contiguous region per block

    double sum = 0.0, sq = 0.0;

#ifdef HTVG_ASYNC
    // Prime buffer 0 (one async b128 per lane: lanes cover 4KB contiguously).
    __builtin_amdgcn_global_load_async_to_lds_b128(
        HTVG_TO_GBL(ht + base + tid * 4u), HTVG_TO_LDS(&tile[0][tid * 4u]), 0, 0);

    for (unsigned t = 0; t < RED_NTILES; ++t) {
        const unsigned cur = t & 1u;
        if (t + 1u < RED_NTILES) {
            __builtin_amdgcn_global_load_async_to_lds_b128(
                HTVG_TO_GBL(ht + base + (t + 1u) * RED_TILE + tid * 4u),
                HTVG_TO_LDS(&tile[cur ^ 1u][tid * 4u]), 0, 0);
            __builtin_amdgcn_s_wait_asynccnt(1);   // current tile landed, next in flight
        } else {
            __builtin_amdgcn_s_wait_asynccnt(0);
        }
        asm volatile("" ::: "memory");             // LDS was written behind the compiler
        const v4f v = *(const v4f*)&tile[cur][tid * 4u];   // ds_load_b128
        sum += (double)v.x + (double)v.y + (double)v.z + (double)v.w;
        sq  += (double)v.x * (double)v.x + (double)v.y * (double)v.y +
               (double)v.z * (double)v.z + (double)v.w * (double)v.w;
    }
#else
    for (unsigned t = 0; t < RED_NTILES; ++t) {
        const v4f v = __builtin_nontemporal_load(
            (const v4f*)(ht + base + t * RED_TILE + tid * 4u));
        sum += (double)v.x + (double)v.y + (double)v.z + (double)v.w;
        sq  += (double)v.x * (double)v.x + (double)v.y * (double)v.y +
               (double)v.z * (double)v.z + (double)v.w * (double)v.w;
    }
#endif

    s_sum[tid] = sum;
    s_sq[tid]  = sq;
    __syncthreads();
    for (unsigned off = RED_BLOCK / 2u; off > 0u; off >>= 1u) {
        if (tid < off) {
            s_sum[tid] += s_sum[tid + off];
            s_sq[tid]  += s_sq[tid + off];
        }
        __syncthreads();
    }
    if (tid == 0) {
        atomicAdd(&ws[0], s_sum[0]);
        atomicAdd(&ws[1], s_sq[0]);
    }
}

// ---------------------------------------------------------------------------
// Kernel 2: main gather + activation + quat->cov + store.
// ---------------------------------------------------------------------------
__device__ __forceinline__ float htvg_sigmoid(float x) {
    return __frcp_rn(1.0f + __expf(-x));
}

__global__ __launch_bounds__(256) void htvg_main(const int*    __restrict__ coords,
                                                 const float*  __restrict__ ht,
                                                 const float*  __restrict__ cam,
                                                 const float*  __restrict__ far_p,
                                                 const int*    __restrict__ vs_p,
                                                 const double* __restrict__ ws,
                                                 int n,
                                                 float* __restrict__ out) {
    const int i = blockIdx.x * blockDim.x + threadIdx.x;
    if (i >= n) return;

    // ---- uniform scalars (compile to s_load / SALU) ----
    const float far_v = far_p[0];
    const float vs    = (float)vs_p[0];
    const float scale_fac = 2.0f * far_v / vs;

    const double M    = (double)REDUCE_N;
    const double s1   = ws[0];
    const double s2   = ws[1];
    const double mean = s1 / M;
    const double var  = (s2 - s1 * s1 / M) / (M - 1.0);
    const float  inv_std = (float)(1.0 / sqrt(var));
    const float  a_norm  = scale_fac * (1.0f / 6.0f) * inv_std;   // (v-mean)*inv_std*sf/6
    const float  b_norm  = -(float)mean * a_norm;

    const float cam0 = cam[0], cam1 = cam[1], cam2 = cam[2];
    const float cbase = -far_v + far_v / vs;   // -far + far/voxel_size

    // ---- per-point: coords (streaming, non-temporal) ----
    const int c0 = __builtin_nontemporal_load(coords + 3 * (size_t)i + 0);
    const int c1 = __builtin_nontemporal_load(coords + 3 * (size_t)i + 1);
    const int c2 = __builtin_nontemporal_load(coords + 3 * (size_t)i + 2);

    // ---- hash index (TABLE_SIZE is 2^22 -> AND mask) ----
    const unsigned h = ((unsigned)c0 * 1u) ^
                       ((unsigned)c1 * 2654435761u) ^
                       ((unsigned)c2 * 805459861u);
    const unsigned idx = h & (TABLE_SIZE_C - 1u);

    // ---- 14 gathers (row 14 "curr" is unused); issue all before consuming ----
    const float* p = ht + idx;
    float f[14];
#pragma unroll
    for (int r = 0; r < 14; ++r) {
        f[r] = p[(size_t)r * TABLE_SIZE_C];
    }

    // ---- activations ----
    const float d0 = f[0] * a_norm + b_norm;
    const float d1 = f[1] * a_norm + b_norm;
    const float d2 = f[2] * a_norm + b_norm;

    const float qr = f[3], qx = f[4], qy = f[5], qz = f[6];

    const float sc0 = htvg_sigmoid(f[7])  * scale_fac;
    const float sc1 = htvg_sigmoid(f[8])  * scale_fac;
    const float sc2 = htvg_sigmoid(f[9])  * scale_fac;

    const float sh0 = htvg_sigmoid(f[10]);
    const float sh1 = htvg_sigmoid(f[11]);
    const float sh2 = htvg_sigmoid(f[12]);

    const float op  = htvg_sigmoid(f[13] - 4.0f);

    // ---- means = dmeans + voxel center ----
    const float m0 = d0 + (float)c0 * scale_fac + cam0 + cbase;
    const float m1 = d1 + (float)c1 * scale_fac + cam1 + cbase;
    const float m2 = d2 + (float)c2 * scale_fac + cam2 + cbase;

    // ---- quaternion -> rotation matrix ----
    const float qn = qr * qr + qx * qx + qy * qy + qz * qz;
    const float rn = __frsqrt_rn(qn);
    const float r  = qr * rn, x = qx * rn, y = qy * rn, z = qz * rn;

    const float xx = x * x, yy = y * y, zz = z * z;
    const float xy = x * y, xz = x * z, yz = y * z;
    const float rx = r * x, ry = r * y, rz = r * z;

    const float R00 = 1.0f - 2.0f * (yy + zz);
    const float R01 = 2.0f * (xy - rz);
    const float R02 = 2.0f * (xz + ry);
    const float R10 = 2.0f * (xy + rz);
    const float R11 = 1.0f - 2.0f * (xx + zz);
    const float R12 = 2.0f * (yz - rx);
    const float R20 = 2.0f * (xz - ry);
    const float R21 = 2.0f * (yz + rx);
    const float R22 = 1.0f - 2.0f * (xx + yy);

    // ---- cov = L L^T with L = R * diag(s); cov[i][k] = sum_j R[i][j]R[k][j] s_j^2 ----
    const float t0 = sc0 * sc0, t1 = sc1 * sc1, t2 = sc2 * sc2;
    const float c00 = R00 * R00 * t0 + R01 * R01 * t1 + R02 * R02 * t2;
    const float c01 = R00 * R10 * t0 + R01 * R11 * t1 + R02 * R12 * t2;
    const float c02 = R00 * R20 * t0 + R01 * R21 * t1 + R02 * R22 * t2;
    const float c11 = R10 * R10 * t0 + R11 * R11 * t1 + R12 * R12 * t2;
    const float c12 = R10 * R20 * t0 + R11 * R21 * t1 + R12 * R22 * t2;
    const float c22 = R20 * R20 * t0 + R21 * R21 * t1 + R22 * R22 * t2;

    // ---- output row: [m0 m1 m2 | c00 c01 c02 c10 c11 c12 c20 c21 c22 | sh0 sh1 sh2 | op]
    // 64 bytes per point, 16B aligned -> 4x global_store_b128 non-temporal.
    // Symmetric entries: c10=c01, c20=c02, c21=c12.
    v4f* o = (v4f*)(out + (size_t)i * 16);
    v4f v0 = { m0,  m1,  m2,  c00 };
    v4f v1 = { c01, c02, c01, c11 };
    v4f v2 = { c12, c02, c12, c22 };
    v4f v3 = { sh0, sh1, sh2, op  };
    __builtin_nontemporal_store(v0, o + 0);
    __builtin_nontemporal_store(v1, o + 1);
    __builtin_nontemporal_store(v2, o + 2);
    __builtin_nontemporal_store(v3, o + 3);
}

// ---------------------------------------------------------------------------
// Host entry
// ---------------------------------------------------------------------------
extern "C" void kernel_launch(void* const* d_in, const int* in_sizes, int n_in,
                              void* d_out, int out_size, void* d_ws, size_t ws_size,
                              hipStream_t stream) {
    const int*   coords = (const int*)  d_in[0];
    const float* ht     = (const float*)d_in[1];
    const float* cam    = (const float*)d_in[2];
    const float* far_p  = (const float*)d_in[3];
    const int*   vs_p   = (const int*)  d_in[4];
    double* ws  = (double*)d_ws;
    float*  out = (float*) d_out;

    const int n = in_sizes[0] / 3;   // number of points

    htvg_zero_ws<<<1, 32, 0, stream>>>(ws);
    htvg_reduce<<<RED_GRID, RED_BLOCK, 0, stream>>>(ht, ws);
    htvg_main<<<(n + 255) / 256, 256, 0, stream>>>(coords, ht, cam, far_p, vs_p, ws, n, out);
}